// TripletFeatures_31018253811800
// MI455X (gfx1250) — compile-verified
//
#include <hip/hip_runtime.h>
#include <stdint.h>

// ---------------- compile-time index set (mirrors the Python _get_index) ----
#define MDIM 41
#define HALF 20
#define ROWF (2 * MDIM) // 82 floats per batch per array (M rows x 2 comps)

constexpr int count_idx() {
    int c = 0;
    for (int m = -HALF; m <= HALF; ++m)
        for (int n = -HALF; n <= HALF; ++n) {
            int mn = m * n; int amn = mn < 0 ? -mn : mn;
            int am = m < 0 ? -m : m; int an = n < 0 ? -n : n;
            if (amn <= HALF && n >= m && (am + an) <= HALF) ++c;
        }
    return c;
}
constexpr int K = count_idx(); // 173

struct PackTab { int v[K]; };
constexpr PackTab make_pack() {
    PackTab t{};
    int c = 0;
    for (int m = -HALF; m <= HALF; ++m)
        for (int n = -HALF; n <= HALF; ++n) {
            int mn = m * n; int amn = mn < 0 ? -mn : mn;
            int am = m < 0 ? -m : m; int an = n < 0 ? -n : n;
            if (amn <= HALF && n >= m && (am + an) <= HALF) {
                int uo = (HALF + m) * 2;          // float offset of U row
                int vo = (HALF + n) * 2;          // float offset of V row
                int wo = (HALF + m + n) * 2;      // float offset of W row
                t.v[c++] = uo | (vo << 8) | (wo << 16) | ((n != m) ? (1 << 24) : 0);
            }
        }
    return t;
}
__constant__ PackTab g_pack = make_pack();

constexpr int B_TILE  = 16;                 // batches per workgroup
constexpr int THREADS = 256;                // 8 wave32
constexpr int TILE_F  = B_TILE * ROWF;      // 1312 floats per array
constexpr int TILE_V4 = TILE_F / 4;         // 328 x b128 chunks per array

__global__ __launch_bounds__(THREADS) void triplet_kernel(
    const float* __restrict__ er, const float* __restrict__ ei,
    float* __restrict__ out, int B) {
    __shared__ __align__(16) float smem[2 * TILE_F];
    float* sre = smem;
    float* sim = smem + TILE_F;

    const int tid = threadIdx.x;
    const long long b0 = (long long)blockIdx.x * B_TILE;
    const long long NF = (long long)B * ROWF;

    // ---- stage inputs: gfx1250 async global->LDS (ASYNCcnt-tracked DMA) ----
    // 16-batch slab of each array is contiguous: b0*328 bytes (16B aligned).
    for (int c = tid; c < 2 * TILE_V4; c += THREADS) {
        const int arr = (c >= TILE_V4);
        const int cc  = arr ? (c - TILE_V4) : c;
        const long long gbase = b0 * ROWF + (long long)cc * 4;
        if (gbase + 4 <= NF) {
            const float* gp = (arr ? ei : er) + gbase;
            // low 32 bits of a flat shared pointer == LDS byte offset
            uint32_t lds = (uint32_t)(size_t)((arr ? sim : sre) + cc * 4);
            asm volatile("global_load_async_to_lds_b128 %0, %1, off"
                         :: "v"(lds), "v"((unsigned long long)(size_t)gp)
                         : "memory");
        }
    }
    asm volatile("s_wait_asynccnt 0" ::: "memory");
    __syncthreads();

    // ---- compute: one (batch, k) pair per thread-iteration ----
    long long rem = (long long)B - b0;
    const int nb = rem < (long long)B_TILE ? (int)rem : B_TILE;
    const int total = nb * K;
    float2* o = (float2*)out;

    for (int idx = tid; idx < total; idx += THREADS) {
        const int bl = idx / K;
        const int k  = idx - bl * K;
        const int pk = g_pack.v[k];
        const int uo = pk & 0xFF, vo = (pk >> 8) & 0xFF, wo = (pk >> 16) & 0xFF;
        const float symf = (float)((pk >> 24) & 1);

        const float* re = sre + bl * ROWF;
        const float* im = sim + bl * ROWF;
        const float u0r = re[uo],     u0i = im[uo];
        const float u1r = re[uo + 1], u1i = im[uo + 1];
        const float v0r = re[vo],     v0i = im[vo];
        const float v1r = re[vo + 1], v1i = im[vo + 1];
        const float w0r = re[wo],     w0i = im[wo];
        const float w1r = re[wo + 1], w1i = im[wo + 1];

        // s_mn = u0*conj(w0) + u1*conj(w1)
        const float smr = u0r * w0r + u0i * w0i + u1r * w1r + u1i * w1i;
        const float smi = u0i * w0r - u0r * w0i + u1i * w1r - u1r * w1i;
        // s_nm (zeroed when m == n, branch-free)
        const float snr = symf * (v0r * w0r + v0i * w0i + v1r * w1r + v1i * w1i);
        const float sni = symf * (v0i * w0r - v0r * w0i + v1i * w1r - v1r * w1i);

        // out[b,c,k] = s_mn*v[c] + s_nm*u[c]
        const float f0r = smr * v0r - smi * v0i + snr * u0r - sni * u0i;
        const float f0i = smr * v0i + smi * v0r + snr * u0i + sni * u0r;
        const float f1r = smr * v1r - smi * v1i + snr * u1r - sni * u1i;
        const float f1i = smr * v1i + smi * v1r + snr * u1i + sni * u1r;

        const long long b = b0 + bl;
        o[(b * 2 + 0) * K + k] = make_float2(f0r, f0i); // coalesced b64 stores
        o[(b * 2 + 1) * K + k] = make_float2(f1r, f1i);
    }
}

extern "C" void kernel_launch(void* const* d_in, const int* in_sizes, int n_in,
                              void* d_out, int out_size, void* d_ws, size_t ws_size,
                              hipStream_t stream) {
    const float* er = (const float*)d_in[0];
    const float* ei = (const float*)d_in[1];
    float* out = (float*)d_out;
    const int B = in_sizes[0] / ROWF; // (B, 41, 2) float32
    const int grid = (B + B_TILE - 1) / B_TILE;
    triplet_kernel<<<grid, THREADS, 0, stream>>>(er, ei, out, B);
}